// GMultiHeadedAttention_45131516346459
// MI455X (gfx1250) — compile-verified
//
#include <hip/hip_runtime.h>

// out = value @ Wv + bv + key
// (softmax over a size-1 axis == 1, so the attention collapses to the V
//  projection plus residual; q/k/U are mathematically dead.)

typedef __attribute__((ext_vector_type(2))) float v2f;
typedef __attribute__((ext_vector_type(8))) float v8f;
typedef __attribute__((ext_vector_type(4))) int   v4i;

#define DM    1024     // d_model (N and K)
#define BM    128      // block tile M
#define BN    64       // block tile N
#define BK    32       // K chunk staged in LDS
#define LDA   36       // A LDS row stride (floats): conflict-free b64 frags, 16B rows
#define LDBT  34       // B^T LDS row stride (floats): conflict-free b64 frags

#if defined(__has_builtin)
#if __has_builtin(__builtin_amdgcn_global_load_async_to_lds_b128)
#define HAVE_ASYNC_LDS 1
#endif
#endif

#ifdef HAVE_ASYNC_LDS
typedef __attribute__((address_space(1))) v4i* gvec_ptr;   // global int4*
typedef __attribute__((address_space(3))) v4i* lvec_ptr;   // LDS    int4*
#endif

__device__ __forceinline__ void wait_async_zero() {
#if defined(__has_builtin) && __has_builtin(__builtin_amdgcn_s_wait_asynccnt)
    __builtin_amdgcn_s_wait_asynccnt(0);
#else
    asm volatile("s_wait_asynccnt 0x0" ::: "memory");
#endif
}

__global__ __launch_bounds__(256)
void vproj_residual_wmma(const float* __restrict__ V,    // value  [2048 x 1024]
                         const float* __restrict__ Wv,   // weights[1024 x 1024]
                         const float* __restrict__ bv,   // bias   [1024]
                         const float* __restrict__ K,    // key (residual) [2048 x 1024]
                         float* __restrict__ out)        // [2048 x 1024]
{
    __shared__ __align__(16) float As[2][BM * LDA];   // 2 x 18432 B
    __shared__ __align__(16) float Bt[2][BN * LDBT];  // 2 x  8704 B (transposed: [n][k])

    const int tid   = threadIdx.x;
    const int lane  = tid & 31;
    const int wid   = tid >> 5;
    const int lm    = lane & 15;          // M (A) / N (B,C) index within 16
    const int khalf = (lane >> 4) << 1;   // 0 or 2: which K-pair this half-wave holds

    const int m0 = blockIdx.y * BM;
    const int n0 = blockIdx.x * BN;

    // 8 waves arranged 4 (M) x 2 (N); each wave computes a 32x32 tile
    const int wm = (wid >> 1) * 32;
    const int wc = (wid & 1) * 32;

    v8f acc[2][2];
    const v8f vzero = {0.f, 0.f, 0.f, 0.f, 0.f, 0.f, 0.f, 0.f};
    acc[0][0] = vzero; acc[0][1] = vzero; acc[1][0] = vzero; acc[1][1] = vzero;

    // global->LDS staging coordinates (float4 per thread)
    const int aCol4 = (tid & 7) * 4;     // 0..28
    const int aRow  = tid >> 3;          // 0..31  (x4 passes -> 128 rows)
    const int bCol4 = (tid & 15) * 4;    // 0..60
    const int bRow  = tid >> 4;          // 0..15  (x2 passes -> 32 rows)

    // ---------------- prologue: stage K-chunk 0 into buffer 0 ----------------
#ifdef HAVE_ASYNC_LDS
    #pragma unroll
    for (int p = 0; p < 4; ++p) {
        const int r = aRow + p * 32;
        __builtin_amdgcn_global_load_async_to_lds_b128(
            (gvec_ptr)(V + (size_t)(m0 + r) * DM + aCol4),
            (lvec_ptr)(&As[0][r * LDA + aCol4]),
            0, 0);
    }
#else
    #pragma unroll
    for (int p = 0; p < 4; ++p) {
        const int r = aRow + p * 32;
        *(float4*)(&As[0][r * LDA + aCol4]) =
            *(const float4*)(V + (size_t)(m0 + r) * DM + aCol4);
    }
#endif
    #pragma unroll
    for (int p = 0; p < 2; ++p) {
        const int r = bRow + p * 16;
        const float4 d = *(const float4*)(Wv + (size_t)r * DM + n0 + bCol4);
        Bt[0][(bCol4 + 0) * LDBT + r] = d.x;   // transpose scatter
        Bt[0][(bCol4 + 1) * LDBT + r] = d.y;
        Bt[0][(bCol4 + 2) * LDBT + r] = d.z;
        Bt[0][(bCol4 + 3) * LDBT + r] = d.w;
    }
#ifdef HAVE_ASYNC_LDS
    wait_async_zero();
#endif
    __syncthreads();

    // ---------------- main loop: double-buffered K chunks ----------------
    for (int kk = 0; kk < DM; kk += BK) {
        const int cur = (kk >> 5) & 1;
        const int nxt = cur ^ 1;
        const bool more = (kk + BK) < DM;

        float4 bReg0 = {}, bReg1 = {};
#ifndef HAVE_ASYNC_LDS
        float4 aReg[4];
#endif
        if (more) {
            // next A tile: async straight into the other LDS buffer
#ifdef HAVE_ASYNC_LDS
            #pragma unroll
            for (int p = 0; p < 4; ++p) {
                const int r = aRow + p * 32;
                __builtin_amdgcn_global_load_async_to_lds_b128(
                    (gvec_ptr)(V + (size_t)(m0 + r) * DM + (kk + BK) + aCol4),
                    (lvec_ptr)(&As[nxt][r * LDA + aCol4]),
                    0, 0);
            }
#else
            #pragma unroll
            for (int p = 0; p < 4; ++p) {
                const int r = aRow + p * 32;
                aReg[p] = *(const float4*)(V + (size_t)(m0 + r) * DM + (kk + BK) + aCol4);
            }
#endif
            // next B tile: prefetch to registers (transpose-stored after compute)
            bReg0 = *(const float4*)(Wv + (size_t)(kk + BK + bRow) * DM + n0 + bCol4);
            bReg1 = *(const float4*)(Wv + (size_t)(kk + BK + bRow + 16) * DM + n0 + bCol4);
        }

        const float* Ac = &As[cur][0];
        const float* Bc = &Bt[cur][0];
        #pragma unroll
        for (int ks = 0; ks < BK; ks += 4) {
            // A frag: lane holds A[lm][ks+khalf .. +1]      (one ds_load_b64)
            // B frag: lane holds B[ks+khalf .. +1][lm_col]  (one ds_load_b64, transposed LDS)
            v2f a[2], b[2];
            a[0] = *(const v2f*)(Ac + (wm +      lm) * LDA  + ks + khalf);
            a[1] = *(const v2f*)(Ac + (wm + 16 + lm) * LDA  + ks + khalf);
            b[0] = *(const v2f*)(Bc + (wc +      lm) * LDBT + ks + khalf);
            b[1] = *(const v2f*)(Bc + (wc + 16 + lm) * LDBT + ks + khalf);

            #pragma unroll
            for (int mt = 0; mt < 2; ++mt)
                #pragma unroll
                for (int nt = 0; nt < 2; ++nt)
                    acc[mt][nt] = __builtin_amdgcn_wmma_f32_16x16x4_f32(
                        /*neg_a=*/false, a[mt], /*neg_b=*/false, b[nt],
                        /*c_mod=*/(short)0, acc[mt][nt],
                        /*reuse_a=*/false, /*reuse_b=*/false);
        }

        if (more) {
#ifndef HAVE_ASYNC_LDS
            #pragma unroll
            for (int p = 0; p < 4; ++p) {
                const int r = aRow + p * 32;
                *(float4*)(&As[nxt][r * LDA + aCol4]) = aReg[p];
            }
#endif
            const int r0 = bRow;
            const int r1 = bRow + 16;
            Bt[nxt][(bCol4 + 0) * LDBT + r0] = bReg0.x;
            Bt[nxt][(bCol4 + 1) * LDBT + r0] = bReg0.y;
            Bt[nxt][(bCol4 + 2) * LDBT + r0] = bReg0.z;
            Bt[nxt][(bCol4 + 3) * LDBT + r0] = bReg0.w;
            Bt[nxt][(bCol4 + 0) * LDBT + r1] = bReg1.x;
            Bt[nxt][(bCol4 + 1) * LDBT + r1] = bReg1.y;
            Bt[nxt][(bCol4 + 2) * LDBT + r1] = bReg1.z;
            Bt[nxt][(bCol4 + 3) * LDBT + r1] = bReg1.w;
#ifdef HAVE_ASYNC_LDS
            wait_async_zero();
#endif
            __syncthreads();
        }
    }

    // ---- epilogue: + bias + residual; C/D layout: VGPR r -> row r (+8 upper half), col = lane%16
    const int rowHalf = (lane >> 4) << 3;  // 0 or 8
    #pragma unroll
    for (int nt = 0; nt < 2; ++nt) {
        const int col  = n0 + wc + nt * 16 + lm;
        const float bias = bv[col];
        #pragma unroll
        for (int mt = 0; mt < 2; ++mt) {
            const int rbase = m0 + wm + mt * 16 + rowHalf;
            #pragma unroll
            for (int r = 0; r < 8; ++r) {
                const size_t idx = (size_t)(rbase + r) * DM + col;
                out[idx] = acc[mt][nt][r] + K[idx] + bias;
            }
        }
    }
}

extern "C" void kernel_launch(void* const* d_in, const int* in_sizes, int n_in,
                              void* d_out, int out_size, void* d_ws, size_t ws_size,
                              hipStream_t stream) {
    // setup_inputs order: 0 query, 1 key, 2 value, 3 Wq, 4 bq, 5 Wk, 6 bk, 7 Wv, 8 bv, 9 U
    const float* key   = (const float*)d_in[1];
    const float* value = (const float*)d_in[2];
    const float* Wv    = (const float*)d_in[7];
    const float* bv    = (const float*)d_in[8];
    float* out = (float*)d_out;
    (void)in_sizes; (void)n_in; (void)out_size; (void)d_ws; (void)ws_size;

    dim3 grid(DM / BN, 2048 / BM);   // 16 x 16 workgroups
    dim3 block(256);
    vproj_residual_wmma<<<grid, block, 0, stream>>>(value, Wv, bv, key, out);
}